// DETRLoss_69939247448490
// MI455X (gfx1250) — compile-verified
//
#include <hip/hip_runtime.h>
#include <math.h>

#define NB 64
#define NQ 300
#define NT 25
#define NC 92          // NUM_CLASSES + 1
#define EOSW 0.1f

typedef __attribute__((ext_vector_type(2))) float v2f;
typedef __attribute__((ext_vector_type(8))) float v8f;

// GIoU on cxcywh boxes (converted to xyxy internally), per reference formula.
__device__ __forceinline__ float giou_cxcywh(const float4 p, const float4 t) {
    float px1 = p.x - 0.5f * p.z, py1 = p.y - 0.5f * p.w;
    float px2 = p.x + 0.5f * p.z, py2 = p.y + 0.5f * p.w;
    float tx1 = t.x - 0.5f * t.z, ty1 = t.y - 0.5f * t.w;
    float tx2 = t.x + 0.5f * t.z, ty2 = t.y + 0.5f * t.w;
    float a1 = (px2 - px1) * (py2 - py1);
    float a2 = (tx2 - tx1) * (ty2 - ty1);
    float iw = fmaxf(fminf(px2, tx2) - fmaxf(px1, tx1), 0.f);
    float ih = fmaxf(fminf(py2, ty2) - fmaxf(py1, ty1), 0.f);
    float inter = iw * ih;
    float uni = a1 + a2 - inter;
    float iou = inter / uni;
    float cw = fmaxf(fmaxf(px2, tx2) - fminf(px1, tx1), 0.f);
    float ch = fmaxf(fmaxf(py2, ty2) - fminf(py1, ty1), 0.f);
    float ac = cw * ch;
    return iou - (ac - uni) / ac;
}

// ---------------- Stage 1: per-row log-softmax stats (max, log-sum-exp) -----
__global__ void k_stats(const float* __restrict__ logits, float* __restrict__ mx_lse) {
    int idx = blockIdx.x * blockDim.x + threadIdx.x;   // b*NQ + q
    if (idx >= NB * NQ) return;
    const float* row = logits + (size_t)idx * NC;
    float m = -INFINITY;
    for (int c = 0; c < NC; ++c) m = fmaxf(m, row[c]);
    float s = 0.f;
    for (int c = 0; c < NC; ++c) s += expf(row[c] - m);
    mx_lse[2 * idx]     = m;
    mx_lse[2 * idx + 1] = logf(s);
}

// ---------------- Stage 2: cost matrix with WMMA one-hot gather -------------
// cost[b][t][q] = 5*L1 + (-(logit[q,id_t]-mx-lse)) + 2*(-giou)
// logit[q,id_t] computed EXACTLY via V_WMMA_F32_16X16X4_F32 with one-hot B.
__global__ void __launch_bounds__(256) k_cost(const float* __restrict__ logits,
                                              const float* __restrict__ pboxes,
                                              const int*   __restrict__ tlabels,
                                              const float* __restrict__ tboxes,
                                              const float* __restrict__ mx_lse,
                                              float* __restrict__ cost) {
    const int b    = blockIdx.x;
    const int wave = threadIdx.x >> 5;
    const int lane = threadIdx.x & 31;
    const int half = lane >> 4;          // 0: lanes 0-15, 1: lanes 16-31
    const int lrow = lane & 15;
    const int NQT = (NQ + 15) / 16;      // 19 query tiles
    const int NNT = (NT + 15) / 16;      // 2 target tiles
    for (int tile = wave; tile < NQT * NNT; tile += (int)(blockDim.x >> 5)) {
        const int qt = tile / NNT, nt = tile % NNT;
        const int qbase = qt * 16, nbase = nt * 16;
        int qa = qbase + lrow;                   // A-frag row (query)
        if (qa >= NQ) qa = NQ - 1;               // clamp; rows >=NQ never stored
        const int tcol = nbase + lrow;           // B-frag column (target)
        const int labn = (tcol < NT) ? tlabels[b * NT + tcol] : -1;
        const float* arow = logits + (size_t)(b * NQ + qa) * NC;
        v8f acc = {};
        for (int k0 = 0; k0 < NC; k0 += 4) {
            // A 16x4 f32: lanes 0-15 -> K = k0,k0+1 ; lanes 16-31 -> K = k0+2,k0+3
            const int ka = k0 + 2 * half;
            v2f a;  a.x  = arow[ka];  a.y = arow[ka + 1];
            // B 4x16 f32 one-hot: VGPR0 row K=ka, VGPR1 row K=ka+1
            v2f bv; bv.x = (labn == ka)     ? 1.f : 0.f;
                    bv.y = (labn == ka + 1) ? 1.f : 0.f;
            acc = __builtin_amdgcn_wmma_f32_16x16x4_f32(
                false, a, false, bv, (short)0, acc, false, false);
        }
        if (tcol < NT) {
            const float4 tb = ((const float4*)tboxes)[b * NT + tcol];
            #pragma unroll
            for (int r = 0; r < 8; ++r) {
                const int qd = qbase + r + 8 * half;   // D: VGPR r -> rows r, r+8
                if (qd < NQ) {
                    const int row = b * NQ + qd;
                    const float mx = mx_lse[2 * row];
                    const float ls = mx_lse[2 * row + 1];
                    const float cclass = -(acc[r] - mx - ls);
                    const float4 pb = ((const float4*)pboxes)[row];
                    const float l1 = fabsf(pb.x - tb.x) + fabsf(pb.y - tb.y) +
                                     fabsf(pb.z - tb.z) + fabsf(pb.w - tb.w);
                    const float g = giou_cxcywh(pb, tb);
                    cost[(size_t)(b * NT + tcol) * NQ + qd] = 5.f * l1 + cclass - 2.f * g;
                }
            }
        }
    }
}

// ---------------- Stage 3: Jonker-Volgenant assignment (f64, per batch) -----
// Transposed problem (rows = targets n=25, cols = queries m=300), exactly as
// the reference. One block per batch; 300-wide scans parallelized in LDS.
__global__ void __launch_bounds__(512) k_jv(const float* __restrict__ cost,
                                            int* __restrict__ matched) {
    const int b = blockIdx.x;
    const int tid = threadIdx.x;
    __shared__ double v_sh[NQ + 1];
    __shared__ double minv_sh[NQ + 1];
    __shared__ double u_sh[NT + 1];
    __shared__ int    p_sh[NQ + 1];
    __shared__ int    way_sh[NQ + 1];
    __shared__ int    used_sh[NQ + 1];
    __shared__ double rv[512];
    __shared__ int    ri[512];
    __shared__ int    j0_sh, i0_sh, done_sh;
    __shared__ double delta_sh;
    const float* cb = cost + (size_t)b * NT * NQ;

    for (int j = tid; j <= NQ; j += 512) { v_sh[j] = 0.0; p_sh[j] = 0; way_sh[j] = 0; }
    if (tid <= NT) u_sh[tid] = 0.0;
    __syncthreads();

    for (int i = 1; i <= NT; ++i) {
        for (int j = tid; j <= NQ; j += 512) { minv_sh[j] = __builtin_inf(); used_sh[j] = 0; }
        if (tid == 0) { p_sh[0] = i; j0_sh = 0; done_sh = 0; }
        __syncthreads();
        while (true) {
            if (tid == 0) { used_sh[j0_sh] = 1; i0_sh = p_sh[j0_sh]; }
            __syncthreads();
            const int i0 = i0_sh;
            const int j0 = j0_sh;
            const float* crow = cb + (size_t)(i0 - 1) * NQ;
            // relax all free columns
            if (tid < NQ) {
                const int j = tid + 1;
                if (!used_sh[j]) {
                    double cur = (double)crow[j - 1] - u_sh[i0] - v_sh[j];
                    if (cur < minv_sh[j]) { minv_sh[j] = cur; way_sh[j] = j0; }
                }
            }
            __syncthreads();
            // argmin over free columns (first-index tie-break like np.argmin)
            double val = __builtin_inf(); int idx = 0x7fffffff;
            if (tid < NQ) {
                const int j = tid + 1;
                if (!used_sh[j]) { val = minv_sh[j]; idx = j; }
            }
            rv[tid] = val; ri[tid] = idx;
            __syncthreads();
            for (int s = 256; s >= 1; s >>= 1) {
                if (tid < s) {
                    double v2 = rv[tid + s]; int i2 = ri[tid + s];
                    if (v2 < rv[tid] || (v2 == rv[tid] && i2 < ri[tid])) {
                        rv[tid] = v2; ri[tid] = i2;
                    }
                }
                __syncthreads();
            }
            if (tid == 0) { delta_sh = rv[0]; j0_sh = ri[0]; }
            __syncthreads();
            const double delta = delta_sh;
            // dual updates: threads handle j = tid (0..NQ); p-values distinct
            if (tid <= NQ) {
                if (used_sh[tid]) { u_sh[p_sh[tid]] += delta; v_sh[tid] -= delta; }
                else              { minv_sh[tid] -= delta; }
            }
            __syncthreads();
            if (tid == 0) { if (p_sh[j0_sh] == 0) done_sh = 1; }
            __syncthreads();
            if (done_sh) break;
        }
        if (tid == 0) {   // augmenting path (serial, tiny)
            int j0 = j0_sh;
            while (j0) { int j1 = way_sh[j0]; p_sh[j0] = p_sh[j1]; j0 = j1; }
        }
        __syncthreads();
    }
    // p[j] = t+1 means query (j-1) matched to target t
    if (tid < NQ) {
        int pi = p_sh[tid + 1];
        if (pi) matched[b * NT + (pi - 1)] = tid;
    }
}

// ---------------- Stage 4: target-class scatter -----------------------------
__global__ void k_tcinit(int* __restrict__ tclass) {
    int idx = blockIdx.x * blockDim.x + threadIdx.x;
    if (idx < NB * NQ) tclass[idx] = NC - 1;       // "no object" = 91
}
__global__ void k_tcscatter(const int* __restrict__ tlabels,
                            const int* __restrict__ matched,
                            int* __restrict__ tclass) {
    int idx = blockIdx.x * blockDim.x + threadIdx.x;
    if (idx < NB * NT) {
        int b = idx / NT;
        tclass[b * NQ + matched[idx]] = tlabels[idx];
    }
}

// ---------------- Stage 5a: CE partial sums (deterministic tree reduce) -----
__global__ void __launch_bounds__(256) k_ce(const float* __restrict__ logits,
                                            const float* __restrict__ mx_lse,
                                            const int*   __restrict__ tclass,
                                            float* __restrict__ part) {
    __shared__ float r1[256], r2[256];
    const int idx = blockIdx.x * 256 + threadIdx.x;   // grid sized exactly NB*NQ/256
    const int tc = tclass[idx];
    const float w = (tc == NC - 1) ? EOSW : 1.f;
    const float nll = -(logits[(size_t)idx * NC + tc] - mx_lse[2 * idx] - mx_lse[2 * idx + 1]);
    r1[threadIdx.x] = w * nll;
    r2[threadIdx.x] = w;
    __syncthreads();
    for (int s = 128; s >= 1; s >>= 1) {
        if (threadIdx.x < s) { r1[threadIdx.x] += r1[threadIdx.x + s]; r2[threadIdx.x] += r2[threadIdx.x + s]; }
        __syncthreads();
    }
    if (threadIdx.x == 0) { part[2 * blockIdx.x] = r1[0]; part[2 * blockIdx.x + 1] = r2[0]; }
}

// ---------------- Stage 5b: matched-pair losses + final combine -------------
__global__ void __launch_bounds__(256) k_final(const float* __restrict__ pboxes,
                                               const float* __restrict__ tboxes,
                                               const int*   __restrict__ matched,
                                               const float* __restrict__ part,
                                               float* __restrict__ out) {
    __shared__ float r1[256], r2[256];
    __shared__ float ceS, wS;
    const int tid = threadIdx.x;
    float s1 = 0.f, s2 = 0.f;
    for (int i = tid; i < (NB * NQ) / 256; i += 256) { s1 += part[2 * i]; s2 += part[2 * i + 1]; }
    r1[tid] = s1; r2[tid] = s2;
    __syncthreads();
    for (int s = 128; s >= 1; s >>= 1) {
        if (tid < s) { r1[tid] += r1[tid + s]; r2[tid] += r2[tid + s]; }
        __syncthreads();
    }
    if (tid == 0) { ceS = r1[0]; wS = r2[0]; }
    __syncthreads();

    float l1s = 0.f, gis = 0.f;
    for (int p = tid; p < NB * NT; p += 256) {
        const int b = p / NT;
        const int q = matched[p];
        const float4 pb = ((const float4*)pboxes)[b * NQ + q];
        const float4 tb = ((const float4*)tboxes)[p];
        l1s += fabsf(pb.x - tb.x) + fabsf(pb.y - tb.y) + fabsf(pb.z - tb.z) + fabsf(pb.w - tb.w);
        gis += 1.f - giou_cxcywh(pb, tb);
    }
    r1[tid] = l1s; r2[tid] = gis;
    __syncthreads();
    for (int s = 128; s >= 1; s >>= 1) {
        if (tid < s) { r1[tid] += r1[tid + s]; r2[tid] += r2[tid + s]; }
        __syncthreads();
    }
    if (tid == 0) {
        const float inv_nt = 1.f / (float)(NB * NT);
        const float lce = 2.0f * (ceS / wS);
        const float lbb = 2.5f * (r1[0] * inv_nt);
        const float lgi = 2.0f * (r2[0] * inv_nt);
        out[0] = lce; out[1] = lbb; out[2] = lgi; out[3] = lce + lbb + lgi;
    }
}

extern "C" void kernel_launch(void* const* d_in, const int* in_sizes, int n_in,
                              void* d_out, int out_size, void* d_ws, size_t ws_size,
                              hipStream_t stream) {
    const float* logits  = (const float*)d_in[0];   // [B,Q,92]
    const float* pboxes  = (const float*)d_in[1];   // [B,Q,4]
    const int*   tlabels = (const int*)  d_in[2];   // [B,T]
    const float* tboxes  = (const float*)d_in[3];   // [B,T,4]
    float* out = (float*)d_out;                     // [4] f32

    // workspace layout (~2.2 MB total)
    float* mx_lse  = (float*)d_ws;                            // 2*B*Q floats
    float* cost    = mx_lse + 2 * NB * NQ;                    // B*T*Q floats
    int*   matched = (int*)(cost + (size_t)NB * NT * NQ);     // B*T ints
    int*   tclass  = matched + NB * NT;                       // B*Q ints
    float* part    = (float*)(tclass + NB * NQ);              // 2 * (B*Q/256) floats

    k_stats    <<<(NB * NQ + 255) / 256, 256, 0, stream>>>(logits, mx_lse);
    k_cost     <<<NB, 256, 0, stream>>>(logits, pboxes, tlabels, tboxes, mx_lse, cost);
    k_jv       <<<NB, 512, 0, stream>>>(cost, matched);
    k_tcinit   <<<(NB * NQ + 255) / 256, 256, 0, stream>>>(tclass);
    k_tcscatter<<<(NB * NT + 255) / 256, 256, 0, stream>>>(tlabels, matched, tclass);
    k_ce       <<<(NB * NQ) / 256, 256, 0, stream>>>(logits, mx_lse, tclass, part);
    k_final    <<<1, 256, 0, stream>>>(pboxes, tboxes, matched, part, out);
}